// ChebyGraphConvolution_32186484916413
// MI455X (gfx1250) — compile-verified
//
#include <hip/hip_runtime.h>
#include <hip/hip_bf16.h>

// ---------------------------------------------------------------------------
// Chebyshev graph convolution for MI455X (gfx1250, wave32).
//   out = sum_k SpMM(A_k, x @ W_k) + bias
//
// Roofline: scatter dominates (~2 GB L2/HBM traffic); GEMM is ~6.5 GFLOP.
// - GEMM: V_WMMA_F32_16X16X4_F32, one wave per 16-row M-tile computing all
//   8 N-tiles (A loaded once per K-step, W_k staged in LDS, B via
//   ds_load_2addr_stride64).
// - Scatter: fused over all 4 Chebyshev degrees -> ONE f32-atomic pass
//   (4x less RMW traffic than per-degree scatter). Falls back to per-degree
//   if d_ws can't hold 4 support buffers.
// ---------------------------------------------------------------------------

typedef __attribute__((ext_vector_type(2))) float v2f;
typedef __attribute__((ext_vector_type(8))) float v8f;

#define F_DIM 128   // F_IN == F_OUT == 128

// out[i] = bias[i % 128]
__global__ __launch_bounds__(256) void cheby_init_bias(
    float* __restrict__ out, const float* __restrict__ bias, int total) {
  int i = blockIdx.x * 256 + threadIdx.x;
  if (i < total) out[i] = bias[i & (F_DIM - 1)];
}

// support = x @ W  for one Chebyshev degree.
// Block = 256 threads = 8 waves; wave w owns M-tile (blockIdx.x*8 + w) and
// computes all 8 N-tiles (N=128). A tile is loaded from global once per
// K-step and reused across the 8 WMMAs; B comes from the LDS-staged W.
__global__ __launch_bounds__(256) void cheby_gemm_xw(
    const float* __restrict__ x, const float* __restrict__ w,
    float* __restrict__ support, int n_mtiles) {
  __shared__ float wl[F_DIM * F_DIM];  // 64 KB

  {  // cooperative, coalesced copy of W into LDS
    const float4* __restrict__ w4 = (const float4*)w;
    float4* wl4 = (float4*)wl;
    #pragma unroll
    for (int i = threadIdx.x; i < (F_DIM * F_DIM) / 4; i += 256)
      wl4[i] = w4[i];
  }
  __syncthreads();

  const int lane  = threadIdx.x & 31;
  const int wave  = threadIdx.x >> 5;
  const int mtile = blockIdx.x * 8 + wave;
  if (mtile >= n_mtiles) return;        // wave-uniform: EXEC stays all-1s

  const int m0   = mtile * 16;
  const int mrow = lane & 15;           // M (A) / N (B,C) sub-index
  const int koff = (lane >> 4) * 2;     // K pair selector per ISA layout

  const float* __restrict__ xrow = x + (size_t)(m0 + mrow) * F_DIM;

  v8f c[8] = {};                        // 8 N-tiles of 16x16 f32 accumulators
  for (int kk = 0; kk < F_DIM; kk += 4) {
    // A: 16x4 f32 tile of x. Lane L: M=L%16; VGPR0=K(koff), VGPR1=K(koff+1).
    v2f a = *(const v2f*)(xrow + kk + koff);
    const float* __restrict__ wr = wl + (kk + koff) * F_DIM + mrow;
    #pragma unroll
    for (int nt = 0; nt < 8; ++nt) {
      // B: 4x16 tile of W (row-major [K][N]); same K striping as A,
      // N = lane%16 striped across lanes within a VGPR.
      v2f b;
      b.x = wr[nt * 16];
      b.y = wr[nt * 16 + F_DIM];
      c[nt] = __builtin_amdgcn_wmma_f32_16x16x4_f32(
          /*neg_a=*/false, a, /*neg_b=*/false, b,
          /*c_mod=*/(short)0, c[nt], /*reuse_a=*/false, /*reuse_b=*/false);
    }
  }

  // C/D layout: VGPR r -> M = r + 8*(lane/16), N = lane%16
  const int mbase = m0 + 8 * (lane >> 4);
  #pragma unroll
  for (int nt = 0; nt < 8; ++nt) {
    const int nc = nt * 16 + mrow;
    #pragma unroll
    for (int r = 0; r < 8; ++r)
      support[(size_t)(mbase + r) * F_DIM + nc] = c[nt][r];
  }
}

// Fused scatter over all 4 degrees: one wave per edge, one float4 per lane.
// acc = sum_k vals_k[e] * support_k[col]  -> single atomic pass into out[row].
__global__ __launch_bounds__(256) void cheby_scatter_fused4(
    const float* __restrict__ support,   // [4, N, 128]
    const float* __restrict__ adj_vals,  // [4, E]
    const int* __restrict__ erow, const int* __restrict__ ecol,
    float* __restrict__ out, int n_edges, size_t nf) {
  const int lane = threadIdx.x & 31;
  const int e    = blockIdx.x * 8 + (threadIdx.x >> 5);
  if (e >= n_edges) return;

  const int r    = erow[e];
  const int cidx = ecol[e];
  const size_t coff = (size_t)cidx * F_DIM + lane * 4;

  float4 acc = make_float4(0.f, 0.f, 0.f, 0.f);
  #pragma unroll
  for (int k = 0; k < 4; ++k) {
    const float  v = adj_vals[(size_t)k * n_edges + e];
    const float4 s = *(const float4*)(support + (size_t)k * nf + coff);
    acc.x += v * s.x; acc.y += v * s.y; acc.z += v * s.z; acc.w += v * s.w;
  }

  float* op = out + (size_t)r * F_DIM + lane * 4;
  unsafeAtomicAdd(op + 0, acc.x);
  unsafeAtomicAdd(op + 1, acc.y);
  unsafeAtomicAdd(op + 2, acc.z);
  unsafeAtomicAdd(op + 3, acc.w);
}

// Per-degree scatter (fallback when d_ws can't hold 4 support buffers).
__global__ __launch_bounds__(256) void cheby_scatter(
    const float* __restrict__ support, const float* __restrict__ vals,
    const int* __restrict__ erow, const int* __restrict__ ecol,
    float* __restrict__ out, int n_edges) {
  const int lane = threadIdx.x & 31;
  const int e    = blockIdx.x * 8 + (threadIdx.x >> 5);
  if (e >= n_edges) return;

  const int   r    = erow[e];
  const int   cidx = ecol[e];
  const float v    = vals[e];

  const float4 s = *(const float4*)(support + (size_t)cidx * F_DIM + lane * 4);
  float* op = out + (size_t)r * F_DIM + lane * 4;
  unsafeAtomicAdd(op + 0, v * s.x);
  unsafeAtomicAdd(op + 1, v * s.y);
  unsafeAtomicAdd(op + 2, v * s.z);
  unsafeAtomicAdd(op + 3, v * s.w);
}

extern "C" void kernel_launch(void* const* d_in, const int* in_sizes, int n_in,
                              void* d_out, int out_size, void* d_ws, size_t ws_size,
                              hipStream_t stream) {
  const float* x        = (const float*)d_in[0];  // [N, 128]
  const float* weight   = (const float*)d_in[1];  // [K, 128, 128]
  const float* bias     = (const float*)d_in[2];  // [128]
  const float* adj_vals = (const float*)d_in[3];  // [K, E]
  const int*   erow     = (const int*)d_in[4];    // [E]
  const int*   ecol     = (const int*)d_in[5];    // [E]
  float*       out      = (float*)d_out;          // [N, 128]

  const int N  = in_sizes[0] / F_DIM;             // 50000
  const int KC = in_sizes[1] / (F_DIM * F_DIM);   // 4 Chebyshev degrees
  const int E  = in_sizes[4];                     // 800000 edges

  const size_t NF       = (size_t)N * F_DIM;
  const int    n_mtiles = N / 16;                 // 3125 (N multiple of 16)
  const int    gemm_blk = (n_mtiles + 7) / 8;

  float* support = (float*)d_ws;

  const int total = N * F_DIM;
  cheby_init_bias<<<(total + 255) / 256, 256, 0, stream>>>(out, bias, total);

  const bool fused = (KC == 4) && (ws_size >= 4 * NF * sizeof(float));

  if (fused) {
    for (int k = 0; k < 4; ++k)
      cheby_gemm_xw<<<gemm_blk, 256, 0, stream>>>(
          x, weight + (size_t)k * F_DIM * F_DIM, support + (size_t)k * NF,
          n_mtiles);
    cheby_scatter_fused4<<<(E + 7) / 8, 256, 0, stream>>>(
        support, adj_vals, erow, ecol, out, E, NF);
  } else {
    for (int k = 0; k < KC; ++k) {
      cheby_gemm_xw<<<gemm_blk, 256, 0, stream>>>(
          x, weight + (size_t)k * F_DIM * F_DIM, support, n_mtiles);
      cheby_scatter<<<(E + 7) / 8, 256, 0, stream>>>(
          support, adj_vals + (size_t)k * E, erow, ecol, out, E);
    }
  }
}